// ChamferLoss_1924145348887
// MI455X (gfx1250) — compile-verified
//
#include <hip/hip_runtime.h>
#include <hip/hip_bf16.h>
#include <math.h>

typedef __attribute__((ext_vector_type(2))) float v2f;
typedef __attribute__((ext_vector_type(8))) float v8f;

#define TILE 16
#define ROWT 2                      // A (row) tiles per block -> 32 rows
#define ROWS (TILE * ROWT)
#define WAVES_PER_BLOCK 8
#define BIG 3.402823466e38f

// Fast-math region: lets fminf lower to bare v_min_num_f32 (no canonicalize
// v_max x,x fixups) and keeps the mins eligible for VOPD dual-issue pairing.
#pragma float_control(push)
#pragma float_control(precise, off)

// For each of the 32 X points owned by this block, compute
//   min_j ||x_i - y_j||  over all Ny points of Y,
// then write the sum of the 32 row-min distances to partial[block].
// d^2(i,j) = |x_i|^2 + (|y_j|^2 - 2 x_i.y_j); the parenthesized term is a
// 16x16 WMMA tile: A=[x,y,z,1] (16x4 f32), B=[-2yx,-2yy,-2yz,|y|^2] (4x16).
// sqrt / max(.,0) / +row-const are monotone, so they commute with the min
// and stay out of the hot loop.
__launch_bounds__(256)
__global__ void chamfer_nn_rowmin_kernel(const float* __restrict__ X,
                                         const float* __restrict__ Y,
                                         int Nx, int Ny,
                                         float* __restrict__ partial)
{
    const int nrt   = Nx / ROWS;
    const int b     = blockIdx.x / nrt;
    const int itile = blockIdx.x % nrt;    // rows [itile*32, itile*32+32)
    const int tid   = threadIdx.x;
    const int wave  = __builtin_amdgcn_readfirstlane(tid >> 5);   // 0..7
    const int lane  = tid & 31;
    const int lhalf = lane >> 4;           // 0 or 1
    const int lidx  = lane & 15;

    const float* Xb = X + (size_t)b * Nx * 3;
    const float* Yb = Y + (size_t)b * Ny * 3;

    // ---- two A tiles (16x4 f32 each): lanes 0-15 -> M=lane, K={0,1};
    //                                   lanes 16-31 -> M=lane-16, K={2,3}
    v2f a0, a1;
    {
        const int base = itile * ROWS;
        const float* xp0 = Xb + (size_t)(base + lidx) * 3;
        const float* xp1 = xp0 + TILE * 3;
        const float x0 = xp0[0], y0 = xp0[1], z0 = xp0[2];
        const float x1 = xp1[0], y1 = xp1[1], z1 = xp1[2];
        a0.x = lhalf ? z0 : x0;  a0.y = lhalf ? 1.0f : y0;
        a1.x = lhalf ? z1 : x1;  a1.y = lhalf ? 1.0f : y1;
    }

    // running per-lane mins of D = (|y|^2 - 2 x.y), one v8f per A tile
    v8f smin0, smin1;
#pragma unroll
    for (int r = 0; r < 8; ++r) { smin0[r] = BIG; smin1[r] = BIG; }

    const v8f czero = {};

    // scalar trip count: wave-uniform, EXEC stays all-ones (WMMA requirement)
    const int iters = (Ny / TILE) / WAVES_PER_BLOCK;
    const float* yp = Yb + (size_t)(wave * TILE + lidx) * 3;
    const size_t ystep = (size_t)WAVES_PER_BLOCK * TILE * 3;

    for (int it = 0; it < iters; ++it, yp += ystep) {
        // ---- B tile (4x16 f32): lanes 0-15 -> N=lane, K={0,1};
        //                         lanes 16-31 -> N=lane-16, K={2,3}
        const float yx = yp[0], yy = yp[1], yz = yp[2];
        const float y2 = yx * yx + yy * yy + yz * yz;
        v2f bb;
        bb.x = lhalf ? (-2.0f * yz) : (-2.0f * yx);
        bb.y = lhalf ? y2           : (-2.0f * yy);

        // two independent WMMAs share the B operand
        v8f c0 = __builtin_amdgcn_wmma_f32_16x16x4_f32(
            false, a0, false, bb, (short)0, czero, false, false);
        v8f c1 = __builtin_amdgcn_wmma_f32_16x16x4_f32(
            false, a1, false, bb, (short)0, czero, false, false);

#pragma unroll
        for (int r = 0; r < 8; ++r) {
            smin0[r] = fminf(smin0[r], c0[r]);
            smin1[r] = fminf(smin1[r], c1[r]);
        }
    }

    // Reduce the min over the 16 columns (lanes) within each 16-lane half.
    // D layout: lane half selects rows {r} vs {r+8}; column = lane&15.
#pragma unroll
    for (int r = 0; r < 8; ++r) {
        float v0 = smin0[r], v1 = smin1[r];
#pragma unroll
        for (int m = 1; m <= 8; m <<= 1) {
            v0 = fminf(v0, __shfl_xor(v0, m, 32));
            v1 = fminf(v1, __shfl_xor(v1, m, 32));
        }
        smin0[r] = v0; smin1[r] = v1;
    }

    __shared__ float red[WAVES_PER_BLOCK][ROWS];
    if (lidx == 0) {
#pragma unroll
        for (int r = 0; r < 8; ++r) {
            red[wave][lhalf * 8 + r]        = smin0[r];
            red[wave][TILE + lhalf * 8 + r] = smin1[r];
        }
    }
    __syncthreads();

    // First 32 threads (= wave 0): combine waves, apply |x|^2 + clamp + sqrt,
    // then sum the 32 row distances across the wave.
    if (tid < ROWS) {
        float v = red[0][tid];
#pragma unroll
        for (int w = 1; w < WAVES_PER_BLOCK; ++w) v = fminf(v, red[w][tid]);

        const float* xq = Xb + (size_t)(itile * ROWS + tid) * 3;
        const float qx = xq[0], qy = xq[1], qz = xq[2];
        const float x2 = qx * qx + qy * qy + qz * qz;
        float d = sqrtf(fmaxf(x2 + v, 0.0f));

        d += __shfl_xor(d, 1, 32);
        d += __shfl_xor(d, 2, 32);
        d += __shfl_xor(d, 4, 32);
        d += __shfl_xor(d, 8, 32);
        d += __shfl_xor(d, 16, 32);
        if (tid == 0) partial[blockIdx.x] = d;
    }
}

#pragma float_control(pop)

// Deterministic final reduction of per-block partial sums -> 3 scalars.
// Kept precise (outside the fast-math region).
__launch_bounds__(256)
__global__ void chamfer_finalize_kernel(const float* __restrict__ pc, int nc,
                                        const float* __restrict__ pa, int na,
                                        float invBN, float invBM,
                                        float* __restrict__ out)
{
    __shared__ float sc[256];
    __shared__ float sa[256];
    const int tid = threadIdx.x;
    float c = 0.0f, a = 0.0f;
    for (int i = tid; i < nc; i += 256) c += pc[i];
    for (int i = tid; i < na; i += 256) a += pa[i];
    sc[tid] = c;
    sa[tid] = a;
    __syncthreads();
    for (int s = 128; s > 0; s >>= 1) {
        if (tid < s) { sc[tid] += sc[tid + s]; sa[tid] += sa[tid + s]; }
        __syncthreads();
    }
    if (tid == 0) {
        const float complete = sc[0] * invBN;  // mean over n of min over m
        const float accuracy = sa[0] * invBM;  // mean over m of min over n
        out[0] = accuracy;
        out[1] = complete;
        out[2] = 0.5f * (accuracy + complete);
    }
}

extern "C" void kernel_launch(void* const* d_in, const int* in_sizes, int n_in,
                              void* d_out, int out_size, void* d_ws, size_t ws_size,
                              hipStream_t stream) {
    (void)n_in; (void)out_size; (void)ws_size;

    const float* trgt = (const float*)d_in[0];   // [B, N, 3]
    const float* pred = (const float*)d_in[1];   // [B, M, 3]
    float* out = (float*)d_out;                  // [accuracy, complete, chamfer]

    const int B = 8;
    const int N = in_sizes[0] / (B * 3);         // 4096
    const int M = in_sizes[1] / (B * 3);         // 4096

    const int nc = B * (N / ROWS);               // partials for "complete"
    const int na = B * (M / ROWS);               // partials for "accuracy"

    float* pc = (float*)d_ws;                    // [nc]
    float* pa = pc + nc;                         // [na]

    // complete: for each trgt point, min over pred
    chamfer_nn_rowmin_kernel<<<nc, 256, 0, stream>>>(trgt, pred, N, M, pc);
    // accuracy: for each pred point, min over trgt
    chamfer_nn_rowmin_kernel<<<na, 256, 0, stream>>>(pred, trgt, M, N, pa);

    chamfer_finalize_kernel<<<1, 256, 0, stream>>>(
        pc, nc, pa, na, 1.0f / (float)(B * N), 1.0f / (float)(B * M), out);
}